// HMSRL_35605278884463
// MI455X (gfx1250) — compile-verified
//
#include <hip/hip_runtime.h>

typedef __attribute__((ext_vector_type(2))) float v2f;
typedef __attribute__((ext_vector_type(8))) float v8f;

#define WAVES_PER_BLOCK 8
#define TILES_PER_WAVE  8   // grid-stride iterations per wave: amortizes resident-W prologue

// out[B,64] = x[B,64] @ W[:, :64]^T + b   (fp32, via V_WMMA_F32_16X16X4_F32)
// One wave -> one 16-row tile of x per iteration, full 64 output columns.
// W fragments (16 kb-steps x 4 nb-tiles) held resident in VGPRs across tiles.
__global__ __launch_bounds__(256) void hmsrl_wmma_f32(
    const float* __restrict__ x, const float* __restrict__ W,
    const float* __restrict__ b, float* __restrict__ out, int tiles_total) {
  const int tid  = threadIdx.x;
  const int lane = tid & 31;
  const int wave = tid >> 5;
  const int m    = lane & 15;   // row (A) / col (B,C) index within tile
  const int half = lane >> 4;   // selects K-pair (A,B) / row-half (C)

  // ---- Resident B fragments: B(k,n) = W[n, k], n = nb*16+m, k = kb*4+half*2+r
  v2f bf[16][4];
#pragma unroll
  for (int kb = 0; kb < 16; ++kb) {
#pragma unroll
    for (int nb = 0; nb < 4; ++nb) {
      bf[kb][nb] = *(const v2f*)(W + (nb * 16 + m) * 128 + kb * 4 + half * 2);
    }
  }

  // ---- Bias (column-only): preload into accumulator init value
  float bias[4];
#pragma unroll
  for (int nb = 0; nb < 4; ++nb) bias[nb] = b[nb * 16 + m];

  const int wave_global = blockIdx.x * WAVES_PER_BLOCK + wave;
  const int stride      = gridDim.x * WAVES_PER_BLOCK;

  for (int t = wave_global; t < tiles_total; t += stride) {  // wave-uniform branch
    const long  row_base = (long)t * 16;
    const float* xt      = x + row_base * 64;

    // Prefetch this wave's next grid-stride tile of x (4KB = 32 lanes * 128B)
    __builtin_prefetch(xt + (long)stride * 1024 + lane * 32, 0, 0);

    v8f acc[4];
#pragma unroll
    for (int nb = 0; nb < 4; ++nb) {
      v8f c;
#pragma unroll
      for (int i = 0; i < 8; ++i) c[i] = bias[nb];
      acc[nb] = c;
    }

#pragma unroll
    for (int kb = 0; kb < 16; ++kb) {
      // A-frag: lane holds x[row_base + m, kb*4 + half*2 + {0,1}]
      v2f a = *(const v2f*)(xt + m * 64 + kb * 4 + half * 2);
#pragma unroll
      for (int nb = 0; nb < 4; ++nb) {
        acc[nb] = __builtin_amdgcn_wmma_f32_16x16x4_f32(
            /*neg_a=*/false, a, /*neg_b=*/false, bf[kb][nb],
            /*c_mod=*/(short)0, acc[nb], /*reuse_a=*/false, /*reuse_b=*/false);
      }
    }

    // C/D layout: VGPR v -> row (v + 8*half), col (nb*16 + m)
    float* ot = out + row_base * 64;
#pragma unroll
    for (int nb = 0; nb < 4; ++nb) {
#pragma unroll
      for (int v = 0; v < 8; ++v) {
        ot[(v + half * 8) * 64 + nb * 16 + m] = acc[nb][v];
      }
    }
  }
}

extern "C" void kernel_launch(void* const* d_in, const int* in_sizes, int n_in,
                              void* d_out, int out_size, void* d_ws, size_t ws_size,
                              hipStream_t stream) {
  const float* x = (const float*)d_in[0];   // [B, 64] fp32
  const float* W = (const float*)d_in[1];   // [64, 128] fp32 (only first 64 cols used)
  const float* b = (const float*)d_in[2];   // [64] fp32
  float*     out = (float*)d_out;           // [B, 64] fp32

  const int D     = 64;
  const long rows = (long)in_sizes[0] / D;  // B = 2,097,152
  const int tiles = (int)(rows / 16);       // 131,072 tiles of 16 rows

  // Launch enough waves to fill the chip, but few enough that each wave
  // iterates ~TILES_PER_WAVE tiles so the resident-W prologue amortizes.
  int blocks = (tiles + WAVES_PER_BLOCK * TILES_PER_WAVE - 1) /
               (WAVES_PER_BLOCK * TILES_PER_WAVE);
  if (blocks < 1) blocks = 1;

  hmsrl_wmma_f32<<<blocks, 256, 0, stream>>>(x, W, b, out, tiles);
}